// StackelbergParamX_30769145708788
// MI455X (gfx1250) — compile-verified
//
#include <hip/hip_runtime.h>

// ---------------------------------------------------------------------------
// StackelbergParamX forward: output == projected X (mask only affects grads).
// Memory-bound (~170-255 MB HBM, ~15us roofline @23.3TB/s). No matmul content
// -> no WMMA (36-wide masked row/col sums are trivial VALU work; WMMA would
// only add LDS relayout traffic). CDNA5 paths used instead:
//   * TDM tensor_load_to_lds  : global->LDS tile staging (1 DMA / matrix)
//   * TDM tensor_store_from_lds: LDS->global result tiles (1 DMA / matrix)
//   * s_wait_tensorcnt, global_prefetch_b8, wave32 codegen
// ---------------------------------------------------------------------------

#define NDIM 36
#define NH   18
#define NN   1296          // 36*36
#define NV4  324           // float4 per matrix
#define NMAT 16384
#define EPSV 1e-8f
#define NEG_SCALE 1.25f

#define K1_BLOCKS 1024
#define K1_THREADS 256
#define K2_THREADS 128
#define K3_BLOCKS 2048
#define K3_THREADS 256

// workspace layout (floats):
//  [0]=s_tl  [1]=s_br  [2..5]=1/d_{tl,br,tr,bl}
//  [WS_PA .. ) : K1_BLOCKS*2 phase-A partials
//  [WS_PB .. ) : NMAT*4 phase-B partials
#define WS_PA 8
#define WS_PB (8 + 2 * K1_BLOCKS)

#ifndef __has_builtin
#define __has_builtin(x) 0
#endif

#if defined(__HIP_DEVICE_COMPILE__) && \
    __has_builtin(__builtin_amdgcn_tensor_load_to_lds) && \
    __has_builtin(__builtin_amdgcn_tensor_store_from_lds) && \
    __has_builtin(__builtin_amdgcn_s_wait_tensorcnt)
#define USE_TDM 1
#else
#define USE_TDM 0
#endif

typedef unsigned int tdm_u32x4 __attribute__((ext_vector_type(4)));
typedef int          tdm_i32x4 __attribute__((ext_vector_type(4)));
typedef int          tdm_i32x8 __attribute__((ext_vector_type(8)));

#if USE_TDM
// Issue one TDM op for a 36x36 f32 tile (5184 B). Descriptor per ISA 8.3/8.4:
// 2D tensor 36x36, data_size=4B, tile == tensor, no padding/iterate/multicast.
// `store` is compile-time constant at call sites.
__device__ __forceinline__ void tdm_tile(unsigned long long ga, unsigned lds_off,
                                         bool store) {
  tdm_u32x4 g0 = {
      1u,                                              // count=1, user D#
      lds_off,                                         // lds_addr (bytes)
      (unsigned)(ga & 0xFFFFFFFFull),                  // global_addr[31:0]
      ((unsigned)(ga >> 32) & 0x1FFFFFFu) | (2u << 30) // global_addr[56:32] | type=2
  };
  tdm_i32x8 g1 = {
      (2 << 16),        // workgroup_mask=0, data_size=2 (4 bytes)
      (NDIM << 16),     // tensor_dim0[15:0] in bits[63:48]
      (NDIM << 16),     // tensor_dim1[15:0] in bits[95:80]
      (NDIM << 16),     // tile_dim0 in bits[127:112]
      NDIM,             // tile_dim1 in bits[143:128]
      NDIM,             // tensor_dim0_stride[31:0]
      (NN << 16),       // tensor_dim1_stride[15:0] (unused for 2D tile; benign)
      0                 // tensor_dim1_stride[47:16]
  };
  tdm_i32x4 z4 = {0, 0, 0, 0};
#if __clang_major__ >= 23
  tdm_i32x8 z8 = {0, 0, 0, 0, 0, 0, 0, 0};
  if (store) __builtin_amdgcn_tensor_store_from_lds(g0, g1, z4, z4, z8, 0);
  else       __builtin_amdgcn_tensor_load_to_lds(g0, g1, z4, z4, z8, 0);
#else
  if (store) __builtin_amdgcn_tensor_store_from_lds(g0, g1, z4, z4, 0);
  else       __builtin_amdgcn_tensor_load_to_lds(g0, g1, z4, z4, 0);
#endif
}
#endif

// ---------------- Phase A: global TL/BR sums of X0 -------------------------
__global__ __launch_bounds__(K1_THREADS)
void k1_block_sums(const float* __restrict__ S, float* __restrict__ partials) {
  __shared__ float red[K1_THREADS * 2];
  const int tid = threadIdx.x;
  const int total = NMAT * NV4;
  const int stride = gridDim.x * K1_THREADS;
  const float4* S4 = (const float4*)S;
  float stl = 0.f, sbr = 0.f;
  for (int q = blockIdx.x * K1_THREADS + tid; q < total; q += stride) {
    __builtin_prefetch(&S4[q + stride], 0, 0);  // -> global_prefetch_b8 (speculative)
    float4 v = S4[q];
    int r = q % NV4;
    int m = r / 9;
    int n0 = (r % 9) * 4;
    bool topm = (m < NH);
    float vv[4] = {v.x, v.y, v.z, v.w};
#pragma unroll
    for (int c = 0; c < 4; ++c) {
      int n = n0 + c;
      if (((n < NH) == topm) && (m != n)) {       // diag blocks minus diagonal
        float x = vv[c] > 0.f ? vv[c] : 0.f;      // relu
        if (topm) stl += x; else sbr += x;
      }
    }
  }
  red[tid * 2 + 0] = stl;
  red[tid * 2 + 1] = sbr;
  __syncthreads();
  for (int off = K1_THREADS / 2; off > 0; off >>= 1) {
    if (tid < off) {
      red[tid * 2 + 0] += red[(tid + off) * 2 + 0];
      red[tid * 2 + 1] += red[(tid + off) * 2 + 1];
    }
    __syncthreads();
  }
  if (tid == 0) {
    partials[blockIdx.x * 2 + 0] = red[0];
    partials[blockIdx.x * 2 + 1] = red[1];
  }
}

__global__ void k_red_a(float* __restrict__ ws) {
  __shared__ float red[256 * 2];
  const int tid = threadIdx.x;
  const float* p = ws + WS_PA;
  float a = 0.f, b = 0.f;
  for (int g = tid; g < K1_BLOCKS; g += 256) { a += p[g * 2]; b += p[g * 2 + 1]; }
  red[tid * 2] = a; red[tid * 2 + 1] = b;
  __syncthreads();
  for (int off = 128; off > 0; off >>= 1) {
    if (tid < off) {
      red[tid * 2] += red[(tid + off) * 2];
      red[tid * 2 + 1] += red[(tid + off) * 2 + 1];
    }
    __syncthreads();
  }
  if (tid == 0) {
    ws[0] = 1.0f / (red[0] + EPSV);   // s_tl (TL_TARGET=1)
    ws[1] = 1.0f / (red[1] + EPSV);   // s_br (BR_TARGET=1)
  }
}

// ---------------- Phase B: per-matrix projection + block partials ----------
__global__ __launch_bounds__(K2_THREADS)
void k2_per_matrix(const float* __restrict__ S, float* __restrict__ out,
                   float* __restrict__ ws) {
  __shared__ __align__(16) float ldsS[NN];   // staging in, reused as result tile out
  __shared__ __align__(16) float ldsX[NN];
  __shared__ float scc[NDIM];
  __shared__ float scr[NDIM];
  __shared__ float red[K2_THREADS * 4];
  const int b = blockIdx.x;
  const int tid = threadIdx.x;
  const float* Sg = S + (size_t)b * NN;

  // ---- stage S tile into LDS: one TDM DMA issued by wave 0 ----------------
#if USE_TDM
  if (tid < 32) {
    tdm_tile((unsigned long long)(uintptr_t)Sg, (unsigned)(uintptr_t)ldsS,
             /*store=*/false);
    __builtin_amdgcn_s_wait_tensorcnt(0);
  }
#else
  for (int i = tid; i < NN; i += K2_THREADS) ldsS[i] = Sg[i];
#endif
  __syncthreads();

  // X1 = scaled, sign-clamped, diag-zeroed (last read of ldsS as input)
  const float s_tl = ws[0], s_br = ws[1];
  for (int i = tid; i < NN; i += K2_THREADS) {
    int m = i / NDIM, n = i % NDIM;
    bool topm = (m < NH), topn = (n < NH);
    float s = ldsS[i];
    float x;
    if (topm == topn) {
      x = (s > 0.f ? s : 0.f);
      if (m == n) x = 0.f;
      x *= topm ? s_tl : s_br;
    } else {
      x = (s < 0.f ? s : 0.f);
    }
    ldsX[i] = x;
  }
  __syncthreads();

  // column pass (sum over rows): sc per column
  if (tid < NDIM) {
    const int n = tid;
    const bool topn = (n < NH);
    float rp = 0.f, rn = 0.f;
    for (int m = 0; m < NDIM; ++m) {
      float v = ldsX[m * NDIM + n];
      if ((m < NH) == topn) rp += v; else rn += v;
    }
    scc[n] = -NEG_SCALE * rp / (rn + EPSV);
  }
  __syncthreads();

  // apply column scale to neg entries, then row pass: sc per row
  if (tid < NDIM) {
    const int m = tid;
    const bool topm = (m < NH);
    float rp = 0.f, rn = 0.f;
    for (int n = 0; n < NDIM; ++n) {
      float v = ldsX[m * NDIM + n];
      if ((n < NH) == topm) {
        rp += v;                              // pos entries untouched by col pass
      } else {
        v *= scc[n];
        ldsX[m * NDIM + n] = v;
        rn += v;
      }
    }
    scr[m] = -NEG_SCALE * rp / (rn + EPSV);
  }
  __syncthreads();

  // row scale + symmetrize + block partial sums; result tile -> ldsS
  float s0 = 0.f, s1 = 0.f, s2 = 0.f, s3 = 0.f;  // tl, br, tr, bl
  float4* Og4 = (float4*)(out + (size_t)b * NN);
  (void)Og4;
  for (int q = tid; q < NV4; q += K2_THREADS) {
    const int m = q / 9;
    const int n0 = (q % 9) * 4;
    const bool topm = (m < NH);
    float res[4];
#pragma unroll
    for (int c = 0; c < 4; ++c) {
      const int n = n0 + c;
      const bool topn = (n < NH);
      const bool isneg = (topm != topn);
      float a = ldsX[m * NDIM + n]; if (isneg) a *= scr[m];
      float t = ldsX[n * NDIM + m]; if (isneg) t *= scr[n];   // same mask at (n,m)
      const float x4 = 0.5f * (a + t);
      res[c] = x4;
      if (topm) { if (topn) s0 += x4; else s2 += x4; }
      else      { if (topn) s3 += x4; else s1 += x4; }
    }
#if USE_TDM
    ((float4*)ldsS)[q] = make_float4(res[0], res[1], res[2], res[3]);
#else
    Og4[q] = make_float4(res[0], res[1], res[2], res[3]);
#endif
  }
  red[tid * 4 + 0] = s0; red[tid * 4 + 1] = s1;
  red[tid * 4 + 2] = s2; red[tid * 4 + 3] = s3;
  __syncthreads();   // also makes the ldsS result tile fully visible
  for (int off = K2_THREADS / 2; off > 0; off >>= 1) {
    if (tid < off) {
#pragma unroll
      for (int k = 0; k < 4; ++k) red[tid * 4 + k] += red[(tid + off) * 4 + k];
    }
    __syncthreads();
  }
  if (tid == 0) {
    float* p = ws + WS_PB + (size_t)b * 4;
    p[0] = red[0]; p[1] = red[1]; p[2] = red[2]; p[3] = red[3];
  }
#if USE_TDM
  // ---- write result tile LDS->global: one TDM store DMA (wave 0) ---------
  // Drained by S_ENDPGM's implicit wait-idle; overlaps with wave retirement.
  if (tid < 32) {
    tdm_tile((unsigned long long)(uintptr_t)(out + (size_t)b * NN),
             (unsigned)(uintptr_t)ldsS, /*store=*/true);
  }
#endif
}

__global__ __launch_bounds__(1024) void k_red_b(float* __restrict__ ws) {
  __shared__ float red[1024 * 4];
  const int tid = threadIdx.x;
  const float* p = ws + WS_PB;
  float s[4] = {0.f, 0.f, 0.f, 0.f};
  for (int b = tid; b < NMAT; b += 1024) {
#pragma unroll
    for (int k = 0; k < 4; ++k) s[k] += p[b * 4 + k];
  }
#pragma unroll
  for (int k = 0; k < 4; ++k) red[tid * 4 + k] = s[k];
  __syncthreads();
  for (int off = 512; off > 0; off >>= 1) {
    if (tid < off) {
#pragma unroll
      for (int k = 0; k < 4; ++k) red[tid * 4 + k] += red[(tid + off) * 4 + k];
    }
    __syncthreads();
  }
  if (tid == 0) {
    const float inv324 = 1.0f / (float)(NH * NH);
    const float dtl =  red[0] * inv324;
    const float dbr =  red[1] * inv324;
    const float dtr = -red[2] * inv324;
    const float dbl = -red[3] * inv324;
    ws[2] = 1.0f / dtl;
    ws[3] = 1.0f / dbr;
    ws[4] = 1.0f / dtr;
    ws[5] = 1.0f / dbl;
  }
}

// ---------------- Phase C: block-wise normalize (in place, L2-resident) ----
__global__ __launch_bounds__(K3_THREADS)
void k3_scale(float* __restrict__ out, const float* __restrict__ ws) {
  const float itl = ws[2], ibr = ws[3], itr = ws[4], ibl = ws[5];
  const int total = NMAT * NV4;
  const int stride = gridDim.x * K3_THREADS;
  float4* O4 = (float4*)out;
  for (int q = blockIdx.x * K3_THREADS + threadIdx.x; q < total; q += stride) {
    __builtin_prefetch(&O4[q + stride], 0, 0);
    float4 v = O4[q];
    const int r = q % NV4;
    const int m = r / 9;
    const int n0 = (r % 9) * 4;
    const bool topm = (m < NH);
    float f[4] = {v.x, v.y, v.z, v.w};
#pragma unroll
    for (int c = 0; c < 4; ++c) {
      const int n = n0 + c;
      const bool topn = (n < NH);
      f[c] *= topm ? (topn ? itl : itr) : (topn ? ibl : ibr);
    }
    O4[q] = make_float4(f[0], f[1], f[2], f[3]);
  }
}

extern "C" void kernel_launch(void* const* d_in, const int* in_sizes, int n_in,
                              void* d_out, int out_size, void* d_ws, size_t ws_size,
                              hipStream_t stream) {
  (void)in_sizes; (void)n_in; (void)out_size; (void)ws_size;
  const float* S = (const float*)d_in[0];
  // d_in[1] (mask_follower) only affects gradients; forward output == Xp.
  float* out = (float*)d_out;
  float* ws  = (float*)d_ws;

  k1_block_sums<<<K1_BLOCKS, K1_THREADS, 0, stream>>>(S, ws + WS_PA);
  k_red_a<<<1, 256, 0, stream>>>(ws);
  k2_per_matrix<<<NMAT, K2_THREADS, 0, stream>>>(S, out, ws);
  k_red_b<<<1, 1024, 0, stream>>>(ws);
  k3_scale<<<K3_BLOCKS, K3_THREADS, 0, stream>>>(out, ws);
}